// QSANNAdapter_80736795230979
// MI455X (gfx1250) — compile-verified
//
#include <hip/hip_runtime.h>
#include <math.h>

typedef float v2f __attribute__((ext_vector_type(2)));
typedef float v8f __attribute__((ext_vector_type(8)));

#define PI_F    3.14159265358979323846f
#define LOG2E_F 1.44269504088896340736f

// ---------------- wave32 helpers ----------------
static __device__ __forceinline__ float shx(float v, int m) { return __shfl_xor(v, m, 32); }
static __device__ __forceinline__ float shl(float v, int src) { return __shfl(v, src, 32); }
static __device__ __forceinline__ float sum16(float v){ v+=shx(v,1); v+=shx(v,2); v+=shx(v,4); v+=shx(v,8); return v; }
static __device__ __forceinline__ float sum32(float v){ v+=shx(v,16); return sum16(v); }

// ---------------- 6-qubit state distributed over one wave32 ----------------
// amplitude index idx = lane*2 + t (t in {0,1}); idx bit0 = t, idx bits 5..1 = lane bits 4..0.
// wire w acts on bit k = 5 - w (wire 0 = MSB, matching the reference).

// Apply RX(x)*RY(y)*RZ(z) on one wire entirely locally by tracking BOTH pair
// halves (a = bit0 side, b = bit1 side) after a single partner exchange.
static __device__ __forceinline__ void wire_rot_pair(
    float& ar, float& ai, float& br, float& bi,
    float cx, float sx, float cy, float sy, float cz, float sz)
{
  // RX: a' = c*a - i*s*b ; b' = -i*s*a + c*b
  float t_ar = cx*ar + sx*bi, t_ai = cx*ai - sx*br;
  float t_br = cx*br + sx*ai, t_bi = cx*bi - sx*ar;
  ar=t_ar; ai=t_ai; br=t_br; bi=t_bi;
  // RY: a' = c*a - s*b ; b' = s*a + c*b
  t_ar = cy*ar - sy*br; t_ai = cy*ai - sy*bi;
  t_br = sy*ar + cy*br; t_bi = sy*ai + cy*bi;
  ar=t_ar; ai=t_ai; br=t_br; bi=t_bi;
  // RZ: a' = (c - i*s)*a ; b' = (c + i*s)*b
  t_ar = cz*ar + sz*ai; t_ai = cz*ai - sz*ar;
  t_br = cz*br - sz*bi; t_bi = cz*bi + sz*br;
  ar=t_ar; ai=t_ai; br=t_br; bi=t_bi;
}

static __device__ __forceinline__ void wire_rot_xlane(
    float sr[2], float si[2], int k,
    float cx, float sx, float cy, float sy, float cz, float sz, int lane)
{
  int m = 1 << (k-1);
  int isB = (lane >> (k-1)) & 1;
  #pragma unroll
  for (int t=0;t<2;++t){
    float pr = shx(sr[t], m), pi2 = shx(si[t], m);   // partner amplitude
    float ar = isB ? pr    : sr[t];
    float ai = isB ? pi2   : si[t];
    float br = isB ? sr[t] : pr;
    float bi = isB ? si[t] : pi2;
    wire_rot_pair(ar, ai, br, bi, cx, sx, cy, sy, cz, sz);
    sr[t] = isB ? br : ar;
    si[t] = isB ? bi : ai;
  }
}

static __device__ __forceinline__ void wire_rot_local(
    float sr[2], float si[2],
    float cx, float sx, float cy, float sy, float cz, float sz)
{
  float ar=sr[0], ai=si[0], br=sr[1], bi=si[1];
  wire_rot_pair(ar, ai, br, bi, cx, sx, cy, sy, cz, sz);
  sr[0]=ar; si[0]=ai; sr[1]=br; si[1]=bi;
}

// The CNOT ring CNOT(0,1),CNOT(1,2),...,CNOT(5,0) is a fixed permutation:
// final[idx] = orig[g1(g2(...g6(idx)))] with g(i) = i ^ (((i>>kc)&1) << kt).
static __device__ __forceinline__ int ring_perm_src(int idx){
  int p = idx;
  #pragma unroll
  for (int i=5; i>=0; --i){
    int kc = 5 - i;
    int kt = 5 - ((i+1)%6);
    p ^= ((p >> kc) & 1) << kt;
  }
  return p;
}

static __device__ __forceinline__ void ring_apply(float sr[2], float si[2],
                                                  int sl0, int st0, int sl1, int st1)
{
  float r00 = shl(sr[0], sl0), r01 = shl(sr[1], sl0);
  float i00 = shl(si[0], sl0), i01 = shl(si[1], sl0);
  float r10 = shl(sr[0], sl1), r11 = shl(sr[1], sl1);
  float i10 = shl(si[0], sl1), i11 = shl(si[1], sl1);
  sr[0] = st0 ? r01 : r00;  si[0] = st0 ? i01 : i00;
  sr[1] = st1 ? r11 : r10;  si[1] = st1 ? i11 : i10;
}

static __device__ __forceinline__ void pqc(const float* __restrict__ w, float sr[2], float si[2],
                                           int lane, int sl0, int st0, int sl1, int st1){
  for (int d=0; d<2; ++d){
    for (int i=0; i<6; ++i){
      const float* wd = w + (d*6+i)*3;
      int k = 5 - i;
      float ax = 0.5f*wd[0], ay = 0.5f*wd[1], az = 0.5f*wd[2];
      float cx=__cosf(ax), sx=__sinf(ax);
      float cy=__cosf(ay), sy=__sinf(ay);
      float cz=__cosf(az), sz=__sinf(az);
      if (k == 0) wire_rot_local(sr, si, cx, sx, cy, sy, cz, sz);
      else        wire_rot_xlane(sr, si, k, cx, sx, cy, sy, cz, sz, lane);
    }
    ring_apply(sr, si, sl0, st0, sl1, st1);
  }
}

static __device__ __forceinline__ void measure_proj_store(
  const float* __restrict__ pjw, const float* __restrict__ pjb,
  const float* __restrict__ gam, const float* __restrict__ bet,
  const float sr[2], const float si[2], int lane, float* __restrict__ dst)
{
  float z[6];
  #pragma unroll
  for (int w=0; w<6; ++w){
    float acc = 0.f;
    #pragma unroll
    for (int t=0;t<2;++t){
      int idx = lane*2 + t;
      float p = sr[t]*sr[t] + si[t]*si[t];
      acc += (((idx >> (5-w)) & 1) ? -p : p);
    }
    z[w] = sum32(acc);   // PauliZ expectation on wire w (all lanes hold it)
  }
  float y[4]; float mean = 0.f;
  #pragma unroll
  for (int j=0;j<4;++j){
    float a = pjb[j];
    #pragma unroll
    for (int w=0;w<6;++w) a += pjw[j*6+w]*z[w];
    y[j]=a; mean += a;
  }
  mean *= 0.25f;
  float var = 0.f;
  #pragma unroll
  for (int j=0;j<4;++j){ float d2 = y[j]-mean; var += d2*d2; }
  var *= 0.25f;
  float inv = 1.0f / sqrtf(var + 1e-5f);
  float o0 = (y[0]-mean)*inv*gam[0] + bet[0];
  float o1 = (y[1]-mean)*inv*gam[1] + bet[1];
  float o2 = (y[2]-mean)*inv*gam[2] + bet[2];
  float o3 = (y[3]-mean)*inv*gam[3] + bet[3];
  if (lane < 4){
    float o = o0;
    o = (lane==1)? o1 : o;
    o = (lane==2)? o2 : o;
    o = (lane==3)? o3 : o;
    dst[lane] = o;
  }
}

// ---------------- kernel 1: per-token quantum front-end (one token / wave) ----------------
__global__ __launch_bounds__(256) void qsann_qk_kernel(
  const float* __restrict__ x,
  const float* __restrict__ ppw, const float* __restrict__ ppb,
  const float* __restrict__ enc_w, const float* __restrict__ q_w, const float* __restrict__ k_w,
  const float* __restrict__ qpw, const float* __restrict__ qpb,
  const float* __restrict__ kpw, const float* __restrict__ kpb,
  const float* __restrict__ gam, const float* __restrict__ bet,
  float* __restrict__ qv, float* __restrict__ kv)
{
  const int lane = threadIdx.x & 31;
  const int tk = blockIdx.x * 8 + (threadIdx.x >> 5);   // token id, 24576 total
  const int bh = tk >> 10;
  const int s  = tk & 1023;

  // precomputed CNOT-ring permutation sources for this lane's two amplitudes
  const int src0 = ring_perm_src(lane*2);
  const int src1 = ring_perm_src(lane*2 + 1);
  const int sl0 = src0 >> 1, st0 = src0 & 1;
  const int sl1 = src1 >> 1, st1 = src1 & 1;

  // lane c holds channel c of this token: x_norm[bh, c, s]
  float xc = x[bh*32768 + lane*1024 + s];

  // 32 -> 6 projection via wave reduction
  float xq[6];
  #pragma unroll
  for (int w=0; w<6; ++w)
    xq[w] = sum32(ppw[w*32 + lane] * xc) + ppb[w];

  float mean = (xq[0]+xq[1]+xq[2]+xq[3]+xq[4]+xq[5]) * (1.0f/6.0f);
  float ss = 0.f;
  #pragma unroll
  for (int w=0;w<6;++w){ float d2 = xq[w]-mean; ss += d2*d2; }
  float stdv = sqrtf(ss * 0.2f) + 1e-6f;   // ddof=1 over 6 values

  float ce[6], se[6];
  #pragma unroll
  for (int w=0;w<6;++w){
    float th = PI_F * tanhf((xq[w]-mean) / stdv);
    ce[w] = __cosf(0.5f*th);
    se[w] = __sinf(0.5f*th);
  }

  // angle-encoded product state (real amplitudes)
  float sr[2], si[2];
  #pragma unroll
  for (int t=0;t<2;++t){
    int idx = lane*2 + t;
    float a = 1.f;
    #pragma unroll
    for (int w=0;w<6;++w) a *= ((idx >> (5-w)) & 1) ? se[w] : ce[w];
    sr[t] = a; si[t] = 0.f;
  }

  pqc(enc_w, sr, si, lane, sl0, st0, sl1, st1);
  float c0=sr[0], c1=sr[1], d0=si[0], d1=si[1];   // save post-enc state

  pqc(q_w, sr, si, lane, sl0, st0, sl1, st1);
  measure_proj_store(qpw, qpb, gam, bet, sr, si, lane, qv + tk*4);

  sr[0]=c0; sr[1]=c1; si[0]=d0; si[1]=d1;
  pqc(k_w, sr, si, lane, sl0, st0, sl1, st1);
  measure_proj_store(kpw, kpb, gam, bet, sr, si, lane, kv + tk*4);
}

// ---------------- kernel 2: RBF flash-attention, all-WMMA ----------------
// Scores <= 0 always (RBF), so softmax needs no running max: p = exp(score).
// log2(e) is folded into the score scaling and exp2 is emitted as a raw
// v_exp_f32 (scores are in [-24, 0], far from the denormal range, so the
// libm denormal fixup is unnecessary).  Row sums of the probability tile
// come from an extra WMMA against an all-ones B operand (P @ 1).
__global__ __launch_bounds__(128) void qsann_attn_kernel(
  const float* __restrict__ qv, const float* __restrict__ kv,
  const float* __restrict__ x,
  const float* __restrict__ p_raw_tau, const float* __restrict__ p_gate,
  float* __restrict__ out)
{
  __shared__ float pbuf[4*16*20];   // per-wave 16x16 P tile, stride 20 (conflict-free)
  const int lane = threadIdx.x & 31;
  const int wslot = threadIdx.x >> 5;
  const int wave = blockIdx.x * 4 + wslot;      // 0..1535
  const int bh = wave >> 6;                     // 24 heads
  const int i0 = (wave & 63) << 4;              // row-tile start within S=1024

  const float4* q4 = (const float4*)(qv + (bh << 12));   // [1024][4] f32 rows
  const float4* k4 = (const float4*)(kv + (bh << 12));
  const float* vb = x + bh*32768;               // V[s][c] = vb[c*1024 + s]
  float* pw = pbuf + wslot*320;

  float rt = p_raw_tau[0];
  float tau = (rt > 20.f) ? rt : log1pf(__expf(rt));   // softplus
  float sc_scale = LOG2E_F / tau;               // score in log2 units
  float gate = p_gate[0];

  const int half = lane >> 4;     // 0: K/M 0..1 | lanes 0-15 ; 1: K 2..3 | lanes 16-31
  const int l16 = lane & 15;
  const int doff = half << 1;

  // A operand (Q tile 16x4, f32 A-layout), pre-scaled by 2*log2e/tau so the
  // score WMMA directly yields log2(p): D = (2/tau) q.k*log2e - (qn+kn)*log2e/tau
  float4 qrow = q4[i0 + l16];
  float s2 = 2.0f * sc_scale;
  v2f aq;
  aq.x = (half ? qrow.z : qrow.x) * s2;
  aq.y = (half ? qrow.w : qrow.y) * s2;

  // per-row -|q|^2*log2e/tau in C-layout row mapping (vgpr v, lane half)
  float qns[8];
  #pragma unroll
  for (int v=0; v<8; ++v){
    float4 qr = q4[i0 + v + half*8];
    qns[v] = -(qr.x*qr.x + qr.y*qr.y + qr.z*qr.z + qr.w*qr.w) * sc_scale;
  }

  v8f oa   = {0.f,0.f,0.f,0.f,0.f,0.f,0.f,0.f};   // output cols 0..15
  v8f ob   = {0.f,0.f,0.f,0.f,0.f,0.f,0.f,0.f};   // output cols 16..31
  v8f lacc = {0.f,0.f,0.f,0.f,0.f,0.f,0.f,0.f};   // per-row softmax denominator
  const v2f ones = {1.f, 1.f};

  for (int j0 = 0; j0 < 1024; j0 += 16){
    // B operand (K^T 4x16, f32 B-layout): lane holds k[j0+l16][doff], [doff+1]
    float4 krow = k4[j0 + l16];
    v2f bk;
    bk.x = half ? krow.z : krow.x;
    bk.y = half ? krow.w : krow.y;
    float kns = -(krow.x*krow.x + krow.y*krow.y + krow.z*krow.z + krow.w*krow.w) * sc_scale;

    // C operand carries the affine part of the RBF score
    v8f cmat;
    #pragma unroll
    for (int v=0; v<8; ++v) cmat[v] = qns[v] + kns;

    v8f sc = __builtin_amdgcn_wmma_f32_16x16x4_f32(false, aq, false, bk,
                                                   (short)0, cmat, false, false);

    // probabilities -> LDS in [row][col]; raw v_exp_f32 (base-2), no fixup
    #pragma unroll
    for (int v=0; v<8; ++v)
      pw[(v + half*8)*20 + l16] = __builtin_amdgcn_exp2f(sc[v]);
    asm volatile("s_wait_dscnt 0" ::: "memory");       // cross-lane RAW via LDS

    // P @ [V | 1]: four K=4 steps; P read back in A-layout, V streamed as B
    #pragma unroll
    for (int kk=0; kk<4; ++kk){
      v2f pa;
      pa.x = pw[l16*20 + kk*4 + doff];
      pa.y = pw[l16*20 + kk*4 + doff + 1];
      const int sb = j0 + kk*4 + doff;
      const float* vc0 = vb + l16*1024;
      const float* vc1 = vb + (16 + l16)*1024;
      v2f bv0; bv0.x = vc0[sb]; bv0.y = vc0[sb+1];
      v2f bv1; bv1.x = vc1[sb]; bv1.y = vc1[sb+1];
      oa   = __builtin_amdgcn_wmma_f32_16x16x4_f32(false, pa, false, bv0,
                                                   (short)0, oa, false, false);
      ob   = __builtin_amdgcn_wmma_f32_16x16x4_f32(false, pa, false, bv1,
                                                   (short)0, ob, false, false);
      lacc = __builtin_amdgcn_wmma_f32_16x16x4_f32(false, pa, false, ones,
                                                   (short)0, lacc, false, false);
    }
    asm volatile("" ::: "memory");   // keep this tile's LDS reads before next stores
  }

  // finalize: divide by softmax denom, apply gate, store out[bh, c, i]
  const int obase = bh*32768;
  #pragma unroll
  for (int v=0; v<8; ++v){
    int row = i0 + v + half*8;
    float invl = gate / lacc[v];
    out[obase + l16*1024 + row]      = oa[v]*invl;
    out[obase + (16+l16)*1024 + row] = ob[v]*invl;
  }
}

// ---------------- host entry ----------------
extern "C" void kernel_launch(void* const* d_in, const int* in_sizes, int n_in,
                              void* d_out, int out_size, void* d_ws, size_t ws_size,
                              hipStream_t stream) {
  (void)in_sizes; (void)n_in; (void)out_size; (void)ws_size;
  const float* x      = (const float*)d_in[0];
  const float* ppw    = (const float*)d_in[1];
  const float* ppb    = (const float*)d_in[2];
  const float* enc_w  = (const float*)d_in[3];
  const float* q_w    = (const float*)d_in[4];
  const float* k_w    = (const float*)d_in[5];
  const float* qpw    = (const float*)d_in[6];
  const float* qpb    = (const float*)d_in[7];
  const float* kpw    = (const float*)d_in[8];
  const float* kpb    = (const float*)d_in[9];
  const float* gam    = (const float*)d_in[10];
  const float* bet    = (const float*)d_in[11];
  const float* rtau   = (const float*)d_in[12];
  const float* gate   = (const float*)d_in[13];
  // d_in[14] = num_heads (fixed at 6 for these shapes)

  float* qv = (float*)d_ws;            // [24576, 4]
  float* kv = qv + 24576*4;            // [24576, 4]

  // 24576 tokens, one per wave32, 8 waves per 256-thread block
  qsann_qk_kernel<<<3072, 256, 0, stream>>>(x, ppw, ppb, enc_w, q_w, k_w,
                                            qpw, qpb, kpw, kpb, gam, bet, qv, kv);
  // 24 heads x 64 row-tiles = 1536 waves, 4 waves per 128-thread block
  qsann_attn_kernel<<<384, 128, 0, stream>>>(qv, kv, x, rtau, gate, (float*)d_out);
}